// Multi_Scale_Region_Distillation_Loss_33414845562965
// MI455X (gfx1250) — compile-verified
//
#include <hip/hip_runtime.h>
#include <stdint.h>

// ---------------- configuration ----------------
#define MAXTPB 256      // max threads per block
#define NBUF   8        // async ring-buffer depth (channels in flight)
#define NCMAX  64       // max classes supported in accumulators
#define PX     2        // pixels per thread (b64 async granularity)

// Per-scale accumulator layout in d_ws (floats):
//   scale s at  ws + s*2*NCMAX :  [0,NCMAX) = sum(sim), [NCMAX,2*NCMAX) = count

// low 32 bits of a flat __shared__ pointer == LDS byte offset (aperture in high bits)
__device__ __forceinline__ uint32_t lds_off(const void* p) {
    return (uint32_t)(uintptr_t)p;
}

// CDNA5 async copy: global (saddr + 32-bit lane offset) -> LDS, 8 bytes/lane.
// Non-temporal: feature maps are streamed exactly once; don't thrash L2.
__device__ __forceinline__ void async_b64(const float* sbase, uint32_t voff_bytes,
                                          uint32_t lds_byte_addr) {
    asm volatile("global_load_async_to_lds_b64 %0, %1, %2 th:TH_LOAD_NT"
                 :: "v"(lds_byte_addr), "v"(voff_bytes), "s"(sbase)
                 : "memory");
}

// ---------------- main per-scale kernel ----------------
// Ring slot layout (per channel): lane t owns 16 bytes at slot_base + 16*t:
//   [0,8) = f pixels (2 floats), [8,16) = fo pixels (2 floats)
// -> read back with ONE ds_load_b128 per channel per lane.
__global__ __launch_bounds__(MAXTPB) void msrd_cos_seg(
    const float* __restrict__ f, const float* __restrict__ fo,
    const int* __restrict__ labels, const int* __restrict__ ncls_p,
    float* __restrict__ acc, int logW, int logTiles, int logRatio, int C)
{
    __shared__ __align__(16) float sbuf[NBUF * MAXTPB * 4]; // 32 KB ring
    __shared__ float csum[NCMAX];
    __shared__ float ccnt[NCMAX];

    const int tid  = (int)threadIdx.x;
    const int tpb  = (int)blockDim.x;
    const int ppb  = tpb * PX;                       // pixels per block
    const int HW   = 1 << (2 * logW);                // W*W (W is a power of two)
    const int tile = (int)blockIdx.x & ((1 << logTiles) - 1);
    const int b    = (int)blockIdx.x >> logTiles;

    if (tid < NCMAX) { csum[tid] = 0.0f; ccnt[tid] = 0.0f; }
    __syncthreads();

    const size_t plane = (size_t)b * (size_t)C * (size_t)HW + (size_t)tile * (size_t)ppb;
    const float* fbase  = f  + plane;   // element index of pixel 0, channel 0
    const float* fobase = fo + plane;

    const uint32_t voff    = (uint32_t)tid * (PX * 4u);      // global lane byte offset
    const uint32_t sstride = (uint32_t)(MAXTPB * 16);        // bytes per ring slot
    const uint32_t lfme    = lds_off(&sbuf[0]) + (uint32_t)tid * 16u;  // f half
    const uint32_t lfome   = lfme + 8u;                      // fo half (same 16B region)

    // ---- prologue: fill the ring (2 async ops per channel, per wave) ----
#pragma unroll
    for (int c = 0; c < NBUF; ++c) {
        async_b64(fbase  + (size_t)c * HW, voff, lfme  + (uint32_t)c * sstride);
        async_b64(fobase + (size_t)c * HW, voff, lfome + (uint32_t)c * sstride);
    }

    float dot0 = 0.f, dot1 = 0.f, nf0 = 0.f, nf1 = 0.f, no0 = 0.f, no1 = 0.f;
    const int slotF = MAXTPB * 4;                    // floats per ring slot

    // ---- steady state: always refill; unroll x8 makes slots compile-time ----
    int c = 0;
#pragma unroll 8
    for (; c < C - NBUF; ++c) {
        // oldest slot's two async loads have landed when <= 2*(NBUF-1) remain
        asm volatile("s_wait_asynccnt 14" ::: "memory");
        const int s = c & (NBUF - 1);
        const float4 v = *(const float4*)&sbuf[s * slotF + 4 * tid];
        // data must be in VGPRs before the async engine overwrites this slot (WAR)
        asm volatile("s_wait_dscnt 0" ::: "memory");
        const int cn = c + NBUF;
        async_b64(fbase  + (size_t)cn * HW, voff, lfme  + (uint32_t)s * sstride);
        async_b64(fobase + (size_t)cn * HW, voff, lfome + (uint32_t)s * sstride);
        dot0 = fmaf(v.x, v.z, dot0);  dot1 = fmaf(v.y, v.w, dot1);
        nf0  = fmaf(v.x, v.x, nf0);   nf1  = fmaf(v.y, v.y, nf1);
        no0  = fmaf(v.z, v.z, no0);   no1  = fmaf(v.w, v.w, no1);
    }

    // ---- drain: all remaining channels are in flight; wait once, no refills ----
    asm volatile("s_wait_asynccnt 0" ::: "memory");
#pragma unroll 8
    for (; c < C; ++c) {
        const int s = c & (NBUF - 1);
        const float4 v = *(const float4*)&sbuf[s * slotF + 4 * tid];
        dot0 = fmaf(v.x, v.z, dot0);  dot1 = fmaf(v.y, v.w, dot1);
        nf0  = fmaf(v.x, v.x, nf0);   nf1  = fmaf(v.y, v.y, nf1);
        no0  = fmaf(v.z, v.z, no0);   no1  = fmaf(v.w, v.w, no1);
    }

    const float EPSV = 1e-8f;
    float sim0, sim1;
    { float n1 = fmaxf(sqrtf(nf0), EPSV), n2 = fmaxf(sqrtf(no0), EPSV); sim0 = dot0 / (n1 * n2); }
    { float n1 = fmaxf(sqrtf(nf1), EPSV), n2 = fmaxf(sqrtf(no1), EPSV); sim1 = dot1 / (n1 * n2); }

    int nc = *ncls_p; if (nc > NCMAX) nc = NCMAX;
    const int p0   = tile * ppb + PX * tid;
    const int wMsk = (1 << logW) - 1;
#pragma unroll
    for (int j = 0; j < PX; ++j) {
        const int p = p0 + j;
        const int h = p >> logW;
        const int w = p & wMsk;
        // nearest-neighbor: src = (dst*512)//H == dst << logRatio exactly (512 % H == 0)
        const int li  = ((b << 9) + (h << logRatio)) * 512 + (w << logRatio);
        const int lab = labels[li];
        const float sv = (j == 0) ? sim0 : sim1;
        if (lab >= 0 && lab < nc) {
            atomicAdd(&csum[lab], sv);
            atomicAdd(&ccnt[lab], 1.0f);
        }
    }
    __syncthreads();
    if (tid < nc) {
        atomicAdd(&acc[tid],         csum[tid]);
        atomicAdd(&acc[NCMAX + tid], ccnt[tid]);
    }
}

// ---------------- helpers ----------------
__global__ void msrd_zero_ws(float* ws) {
    ws[blockIdx.x * blockDim.x + threadIdx.x] = 0.0f;
}

__global__ void msrd_finalize(const float* __restrict__ ws,
                              const int* __restrict__ ncp,
                              const int* __restrict__ nocp,
                              float* __restrict__ out)
{
    if (threadIdx.x == 0 && blockIdx.x == 0) {
        int nc = *ncp; if (nc > NCMAX) nc = NCMAX;
        const int no = *nocp;
        const float wgt[4] = {1.0f, 2.0f, 3.0f, 4.0f};
        float loss = 0.0f;
        for (int s = 0; s < 4; ++s) {
            const float* a = ws + s * (2 * NCMAX);
            for (int c = 0; c < nc; ++c) {
                const float cnt = a[NCMAX + c];
                if (cnt > 0.0f) {
                    const float mean = a[c] / fmaxf(cnt, 1.0f);
                    const float factor = (c == 0) ? (float)no / (float)nc
                                                  : ((c <= no) ? 1.0f : 0.0f);
                    loss += wgt[s] * factor * (1.0f - mean);
                }
            }
        }
        out[0] = loss;
    }
}

// ---------------- entry point ----------------
extern "C" void kernel_launch(void* const* d_in, const int* in_sizes, int n_in,
                              void* d_out, int out_size, void* d_ws, size_t ws_size,
                              hipStream_t stream) {
    // dict order: pseudo_labels, f0, fo0, f1, fo1, f2, fo2, f3, fo3, num_class, num_old_class
    const int*   labels = (const int*)d_in[0];
    const float* fs[4]  = {(const float*)d_in[1], (const float*)d_in[3],
                           (const float*)d_in[5], (const float*)d_in[7]};
    const float* fos[4] = {(const float*)d_in[2], (const float*)d_in[4],
                           (const float*)d_in[6], (const float*)d_in[8]};
    const int* ncp  = (const int*)d_in[9];
    const int* nocp = (const int*)d_in[10];
    float* ws = (float*)d_ws;

    const int B = in_sizes[0] / (512 * 512);

    // zero 4 * 2 * NCMAX = 512 accumulator floats (ws is poisoned, not re-zeroed)
    msrd_zero_ws<<<2, 256, 0, stream>>>(ws);

    const int logWl[4] = {7, 6, 5, 4};               // W = 128, 64, 32, 16
    for (int s = 0; s < 4; ++s) {
        const int logW = logWl[s];
        const int W    = 1 << logW;
        const int HW   = W * W;
        const int C    = in_sizes[1 + 2 * s] / (B * HW);   // 256
        int tpb = HW / PX; if (tpb > MAXTPB) tpb = MAXTPB;
        const int tiles    = HW / (tpb * PX);
        const int logTiles = __builtin_ctz(tiles);
        const int logRatio = 9 - logW;               // 512 / W
        const int grid     = B * tiles;
        msrd_cos_seg<<<grid, tpb, 0, stream>>>(fs[s], fos[s], labels, ncp,
                                               ws + s * 2 * NCMAX,
                                               logW, logTiles, logRatio, C);
    }
    msrd_finalize<<<1, 32, 0, stream>>>(ws, ncp, nocp, (float*)d_out);
}